// TaskGraphLoss_71957882077472
// MI455X (gfx1250) — compile-verified
//
#include <hip/hip_runtime.h>
#include <hip/hip_bf16.h>
#include <stdint.h>

#define NB 32
#define NC 64
#define NT 8192

typedef __attribute__((ext_vector_type(2))) float v2f;
typedef __attribute__((ext_vector_type(8))) float v8f;

// ---------------------------------------------------------------------------
// Kernel 1: channel argmax, streaming. blockIdx.y selects pred (0) / label (1).
// Each thread owns 4 consecutive t (float4 loads -> global_load_b128), loops
// over the 64 channels keeping running max + first-occurrence index.
// ---------------------------------------------------------------------------
__global__ void __launch_bounds__(256) argmax_kernel(const float* __restrict__ pred,
                                                     const float* __restrict__ label,
                                                     uint8_t* __restrict__ idxP,
                                                     uint8_t* __restrict__ idxT) {
    const int tilesPerB = NT / 1024;              // 8 tiles of 1024 t per batch
    const int b    = blockIdx.x / tilesPerB;
    const int tile = blockIdx.x % tilesPerB;
    const int t0   = tile * 1024 + threadIdx.x * 4;

    const float* in   = (blockIdx.y == 0) ? pred : label;
    uint8_t*     outI = (blockIdx.y == 0) ? idxP : idxT;

    const float4* base =
        reinterpret_cast<const float4*>(in + (size_t)b * NC * NT + t0);

    float4 mv = base[0];                           // c = 0
    uchar4 mi = make_uchar4(0, 0, 0, 0);

#pragma unroll 4
    for (int c = 1; c < NC; ++c) {
        float4 v = base[(size_t)c * (NT / 4)];
        if (v.x > mv.x) { mv.x = v.x; mi.x = (uint8_t)c; }
        if (v.y > mv.y) { mv.y = v.y; mi.y = (uint8_t)c; }
        if (v.z > mv.z) { mv.z = v.z; mi.z = (uint8_t)c; }
        if (v.w > mv.w) { mv.w = v.w; mi.w = (uint8_t)c; }
    }
    *reinterpret_cast<uchar4*>(outI + (size_t)b * NT + t0) = mi;
}

// ---------------------------------------------------------------------------
// Kernel 2: per-batch 64x64 transition histograms in LDS (ds_add_u32 atomics),
// excluding self-transitions; written out as f32 counts (deterministic).
// ---------------------------------------------------------------------------
__global__ void __launch_bounds__(256) adjacency_kernel(const uint8_t* __restrict__ idxP,
                                                        const uint8_t* __restrict__ idxT,
                                                        float* __restrict__ cntP,
                                                        float* __restrict__ cntT) {
    __shared__ uint32_t hP[NC * NC];
    __shared__ uint32_t hT[NC * NC];
    const int b = blockIdx.x;

    for (int i = threadIdx.x; i < NC * NC; i += blockDim.x) { hP[i] = 0u; hT[i] = 0u; }
    __syncthreads();

    const uint8_t* ip = idxP + (size_t)b * NT;
    const uint8_t* it = idxT + (size_t)b * NT;
    for (int t = threadIdx.x; t < NT - 1; t += blockDim.x) {
        int c0 = ip[t], c1 = ip[t + 1];
        if (c0 != c1) atomicAdd(&hP[c0 * NC + c1], 1u);
        int d0 = it[t], d1 = it[t + 1];
        if (d0 != d1) atomicAdd(&hT[d0 * NC + d1], 1u);
    }
    __syncthreads();

    for (int i = threadIdx.x; i < NC * NC; i += blockDim.x) {
        cntP[(size_t)b * NC * NC + i] = (float)hP[i];
        cntT[(size_t)b * NC * NC + i] = (float)hT[i];
    }
}

// ---------------------------------------------------------------------------
// Kernel 3: row sums as a matrix op: rowsum(M) = M(64x64) x ones(64x1),
// mapped to V_WMMA_F32_16X16X4_F32 (exact f32). One wave per 16-row slab;
// 16 k-steps of K=4. B = all-ones is VGPR-layout independent.
// A 16x4 f32 layout: lanes 0-15 hold (K=k0, k0+1), lanes 16-31 hold (k0+2, k0+3).
// D 16x16 f32 layout: VGPR r -> row r (lanes 0-15) / row 8+r (lanes 16-31).
// 256 waves total: 2 arrays * 32 batches * 4 slabs.
// ---------------------------------------------------------------------------
__global__ void __launch_bounds__(256) rowsum_kernel(const float* __restrict__ cntT,
                                                     const float* __restrict__ cntP,
                                                     float* __restrict__ rsT,
                                                     float* __restrict__ rsP) {
    const int wave  = (blockIdx.x * blockDim.x + threadIdx.x) >> 5;
    const int lane  = threadIdx.x & 31;
    const int which = wave >> 7;         // 0 = true counts, 1 = pred counts
    const int rem   = wave & 127;
    const int b     = rem >> 2;
    const int slab  = rem & 3;

    const float* M   = ((which == 0) ? cntT : cntP) + (size_t)b * NC * NC + slab * 16 * NC;
    float*       out = ((which == 0) ? rsT  : rsP ) + (size_t)b * NC + slab * 16;

#if defined(__gfx1250__) && __has_builtin(__builtin_amdgcn_wmma_f32_16x16x4_f32)
    const int m    = lane & 15;
    const int koff = (lane < 16) ? 0 : 2;
    v8f acc = {};
    v2f onesB;
    onesB.x = 1.0f;
    onesB.y = 1.0f;
#pragma unroll
    for (int kk = 0; kk < 16; ++kk) {
        const int k0 = kk * 4 + koff;
        v2f a;
        a.x = M[m * NC + k0];
        a.y = M[m * NC + k0 + 1];
        acc = __builtin_amdgcn_wmma_f32_16x16x4_f32(
            /*neg_a=*/false, a, /*neg_b=*/false, onesB,
            /*c_mod=*/(short)0, acc, /*reuse_a=*/false, /*reuse_b=*/false);
    }
    if (lane == 0) {
#pragma unroll
        for (int r = 0; r < 8; ++r) out[r] = acc[r];
    } else if (lane == 16) {
#pragma unroll
        for (int r = 0; r < 8; ++r) out[8 + r] = acc[r];
    }
#else
    if (lane < 16) {
        float s = 0.0f;
        for (int j = 0; j < NC; ++j) s += M[lane * NC + j];
        out[lane] = s;
    }
#endif
}

// ---------------------------------------------------------------------------
// Kernel 4: normalize, BCE + masked MSE, block-wide reduction, scalar out.
// ---------------------------------------------------------------------------
__global__ void __launch_bounds__(1024) loss_kernel(const float* __restrict__ cntT,
                                                    const float* __restrict__ cntP,
                                                    const float* __restrict__ rsT,
                                                    const float* __restrict__ rsP,
                                                    float* __restrict__ out) {
    const int N = NB * NC * NC;           // 131072
    float bce = 0.0f, mse = 0.0f;
    int cnt = 0;

    for (int i = threadIdx.x; i < N; i += blockDim.x) {
        const int row = i >> 6;           // b*NC + r
        const float ta = cntT[i] / (rsT[row] + 1e-8f);
        const float pa = cntP[i] / (rsP[row] + 1e-8f);
        const float y  = (ta != 0.0f) ? 1.0f : 0.0f;
        const float p  = tanhf(pa);
        const float lp  = fmaxf(logf(p), -100.0f);     // log(0)=-inf -> clip to -100
        const float l1p = fmaxf(log1pf(-p), -100.0f);
        bce -= y * lp + (1.0f - y) * l1p;
        if (ta != 0.0f) {
            const float d = pa - ta;
            mse += d * d;
            ++cnt;
        }
    }

    __shared__ float sb[1024];
    __shared__ float sm[1024];
    __shared__ int   sc[1024];
    sb[threadIdx.x] = bce;
    sm[threadIdx.x] = mse;
    sc[threadIdx.x] = cnt;
    __syncthreads();
    for (int s = 512; s > 0; s >>= 1) {
        if ((int)threadIdx.x < s) {
            sb[threadIdx.x] += sb[threadIdx.x + s];
            sm[threadIdx.x] += sm[threadIdx.x + s];
            sc[threadIdx.x] += sc[threadIdx.x + s];
        }
        __syncthreads();
    }
    if (threadIdx.x == 0) {
        const float bceMean = sb[0] / (float)N;
        const float mseMean = (sc[0] > 0) ? sm[0] / (float)sc[0] : 0.0f;
        out[0] = bceMean + mseMean;
    }
}

// ---------------------------------------------------------------------------
extern "C" void kernel_launch(void* const* d_in, const int* in_sizes, int n_in,
                              void* d_out, int out_size, void* d_ws, size_t ws_size,
                              hipStream_t stream) {
    const float* pred  = (const float*)d_in[0];
    const float* label = (const float*)d_in[1];

    uint8_t* ws   = (uint8_t*)d_ws;
    uint8_t* idxP = ws;                                   // NB*NT bytes
    uint8_t* idxT = ws + (size_t)NB * NT;                 // NB*NT bytes
    float*   cntT = (float*)(ws + 2 * (size_t)NB * NT);   // NB*NC*NC f32
    float*   cntP = cntT + (size_t)NB * NC * NC;          // NB*NC*NC f32
    float*   rsT  = cntP + (size_t)NB * NC * NC;          // NB*NC f32
    float*   rsP  = rsT + (size_t)NB * NC;                // NB*NC f32

    dim3 g1(NB * (NT / 1024), 2);
    argmax_kernel<<<g1, 256, 0, stream>>>(pred, label, idxP, idxT);
    adjacency_kernel<<<NB, 256, 0, stream>>>(idxP, idxT, cntP, cntT);
    rowsum_kernel<<<32, 256, 0, stream>>>(cntT, cntP, rsT, rsP);
    loss_kernel<<<1, 1024, 0, stream>>>(cntT, cntP, rsT, rsP, (float*)d_out);
}